// RuchbahSelectiveScan_14963666059311
// MI455X (gfx1250) — compile-verified
//
#include <hip/hip_runtime.h>

#define B_ 4
#define S_ 4096
#define D_ 2048
#define R_ 64

#define DT 128       // d-channels per block
#define SC 32        // timesteps per chunk
#define THREADS 128  // 4 waves (wave32)
#define WPAD 68      // padded row stride for W / delta tiles (R_ + 4)
#define DPAD 132     // padded row stride for decay tile (DT + 4)

typedef __attribute__((ext_vector_type(2))) float v2f;
typedef __attribute__((ext_vector_type(8))) float v8f;

__global__ __launch_bounds__(THREADS)
void RuchbahSelectiveScan_kernel(const float* __restrict__ u,
                                 const float* __restrict__ delta,
                                 const float* __restrict__ W,
                                 const float* __restrict__ bias,
                                 const float* __restrict__ A_log,
                                 const float* __restrict__ Dp,
                                 float* __restrict__ out) {
  __shared__ float Wl[DT][WPAD];       // W tile, row = local d, col = r
  __shared__ float dl[SC][WPAD];       // delta chunk, row = t, col = r
  __shared__ float decay[SC][DPAD];    // decay chunk, row = t, col = local d
  __shared__ float Achan[DT];
  __shared__ float BiasL[DT];

  const int tid  = threadIdx.x;
  const int wave = tid >> 5;
  const int lane = tid & 31;
  const int hf   = lane >> 4;   // which half of the wave (K-split for wmma)
  const int l16  = lane & 15;
  const int d0   = blockIdx.x * DT;
  const int b    = blockIdx.y;

  // ---- preload W tile (DT x R) into LDS (once) ----
  {
    const float4* Wg = (const float4*)(W + (size_t)d0 * R_);
    for (int idx = tid; idx < DT * (R_ / 4); idx += THREADS) {
      int row = idx >> 4;
      int c4  = idx & 15;
      float4 v = Wg[row * (R_ / 4) + c4];
      *(float4*)&Wl[row][c4 * 4] = v;
    }
  }
  // ---- per-channel constants ----
  {
    float Av = -__expf(A_log[d0 + tid]);
    Av = fminf(fmaxf(Av, -10.0f), -1e-6f);
    Achan[tid] = Av;
    BiasL[tid] = bias[d0 + tid];
  }
  const float Dpv = Dp[d0 + tid];
  float state = 0.0f;

  const size_t ubase = ((size_t)b * S_) * D_ + d0;

  for (int s0 = 0; s0 < S_; s0 += SC) {
    // ---- stage delta chunk [SC x R] to LDS (coalesced b128 loads) ----
    {
      const float4* dg = (const float4*)(delta + ((size_t)b * S_ + s0) * R_);
      for (int idx = tid; idx < SC * (R_ / 4); idx += THREADS) {
        int row = idx >> 4;
        int c4  = idx & 15;
        float4 v = dg[row * (R_ / 4) + c4];
        *(float4*)&dl[row][c4 * 4] = v;
      }
    }
    __syncthreads();

    // ---- dt[SC x DT] = delta_chunk[SC x R] * W[DT x R]^T via f32 WMMA ----
    // 16 output tiles of 16x16, 4 per wave, K accumulated 4 at a time.
#pragma unroll
    for (int i = 0; i < 4; ++i) {
      const int tt = wave * 4 + i;
      const int m0 = (tt >> 3) * 16;  // time tile
      const int n0 = (tt & 7) * 16;   // channel tile
      v8f acc = {};
#pragma unroll
      for (int k0 = 0; k0 < R_; k0 += 4) {
        // A: 16x4 f32, lanes 0-15 hold K=k0,k0+1; lanes 16-31 hold K=k0+2,k0+3
        v2f a  = *(const v2f*)&dl[m0 + l16][k0 + 2 * hf];
        // B: 4x16 f32 = W^T slice, same K-split across lane halves
        v2f bm = *(const v2f*)&Wl[n0 + l16][k0 + 2 * hf];
        acc = __builtin_amdgcn_wmma_f32_16x16x4_f32(
            false, a, false, bm, (short)0, acc, false, false);
      }
      // epilogue: softplus + clamps + exp -> decay in LDS
      const int dLoc = n0 + l16;
      const float Av = Achan[dLoc];
      const float bv = BiasL[dLoc];
#pragma unroll
      for (int j = 0; j < 8; ++j) {
        int t = m0 + j + 8 * hf;  // C layout: VGPR j = rows M=j / M=j+8
        float x   = acc[j] + bv;
        float sp  = fmaxf(x, 0.0f) + __logf(1.0f + __expf(-fabsf(x)));
        float dtv = fminf(fmaxf(sp, 1e-6f), 10.0f);
        float dec = fminf(fmaxf(__expf(dtv * Av), 1e-6f), 1.0f);
        decay[t][dLoc] = dec;
      }
    }
    __syncthreads();

    // ---- sequential scan: lane owns channel d0+tid ----
    const float* up = u   + ubase + (size_t)s0 * D_ + tid;
    float*       op = out + ubase + (size_t)s0 * D_ + tid;
#pragma unroll 8
    for (int t = 0; t < SC; ++t) {
      float uu  = up[(size_t)t * D_];          // coalesced, chain-independent
      float dec = decay[t][tid];               // conflict-free LDS read
      state = fminf(fmaxf(fmaf(state, dec, uu), -10000.0f), 10000.0f);
      float fin = fminf(fmaxf(fmaf(uu, Dpv, state), -10000.0f), 10000.0f);
      op[(size_t)t * D_] = fin;                // coalesced store
    }
    // no barrier needed here: next chunk's delta staging touches dl only,
    // and its __syncthreads() orders it against this chunk's decay reads.
  }
}

extern "C" void kernel_launch(void* const* d_in, const int* in_sizes, int n_in,
                              void* d_out, int out_size, void* d_ws, size_t ws_size,
                              hipStream_t stream) {
  const float* u     = (const float*)d_in[0];
  const float* delta = (const float*)d_in[1];
  const float* W     = (const float*)d_in[2];
  const float* bias  = (const float*)d_in[3];
  const float* A_log = (const float*)d_in[4];
  const float* Dp    = (const float*)d_in[5];
  float* out = (float*)d_out;

  dim3 grid(D_ / DT, B_);  // 16 x 4 = 64 blocks, 4 waves each
  RuchbahSelectiveScan_kernel<<<grid, THREADS, 0, stream>>>(
      u, delta, W, bias, A_log, Dp, out);
}